// Encoder_59708635349234
// MI455X (gfx1250) — compile-verified
//
#include <hip/hip_runtime.h>

// ---------------------------------------------------------------------------
// GCN forward for MI455X (gfx1250, wave32).
//  h = x @ W_emb
//  3x { agg = segment_sum(h[src], dst); h = relu(agg@W + b) (+ h) }
//  out = segment_sum(h, node2graph)
// GEMMs use V_WMMA_F32_16X16X4_F32 (full fp32; GEMM cost is negligible in the
// roofline, edge scatter dominates and is L2-resident at 25.6MB per buffer).
// ---------------------------------------------------------------------------

typedef __attribute__((ext_vector_type(2))) float v2f;
typedef __attribute__((ext_vector_type(8))) float v8f;

#define GCN_N_NODES  50000
#define GCN_N_EDGES  800000
#define GCN_N_GRAPHS 500
#define GCN_INP      64
#define GCN_EMB      128

// ---------------------------------------------------------------------------
// GEMM: out[M x 128] = epilogue(A[M x K] @ W[K x 128])
// Grid: M/16 blocks of 256 threads (8 waves). Wave w owns n-tile w (16 cols).
// Each wave: 16x16 f32 tile via K/4 chained V_WMMA_F32_16X16X4_F32.
// A layout (16x4 f32, ISA 7.12.2): lanes 0-15 = M rows, VGPR0 = K even pair
//   member, lane[4] selects K pair {0,1} vs {2,3}.
// B layout (4x16 f32): lanes = N cols, same K striping.
// C/D layout: VGPR r -> M = r + 8*lane[4], N = lane[3:0].
// ---------------------------------------------------------------------------
__global__ __launch_bounds__(256) void gcn_gemm_n128_wmma(
    const float* __restrict__ A, int K,
    const float* __restrict__ W,
    const float* __restrict__ bias,    // nullptr -> no bias
    const float* __restrict__ resid,   // nullptr -> no residual
    float* __restrict__ out,
    int do_relu)
{
    const int wave  = threadIdx.x >> 5;
    const int lane  = threadIdx.x & 31;
    const int m0    = blockIdx.x << 4;      // 16 rows per block
    const int n0    = wave << 4;            // 16 cols per wave (8 waves * 16 = 128)
    const int r15   = lane & 15;
    const int khalf = (lane >> 4) << 1;     // 0 or 2: which K pair this half-wave holds

    const float* Arow = A + (size_t)(m0 + r15) * (size_t)K;

    v8f acc = {};
#pragma unroll 4
    for (int k = 0; k < K; k += 4) {
        v2f a, b;
        a.x = Arow[k + khalf + 0];
        a.y = Arow[k + khalf + 1];
        b.x = W[(size_t)(k + khalf + 0) * GCN_EMB + n0 + r15];
        b.y = W[(size_t)(k + khalf + 1) * GCN_EMB + n0 + r15];
        // (neg_a, A, neg_b, B, c_mod, C, reuse_a, reuse_b)
        acc = __builtin_amdgcn_wmma_f32_16x16x4_f32(
            false, a, false, b, (short)0, acc, false, false);
    }

    const int mbase = m0 + ((lane >> 4) << 3);  // +8 row offset for upper half-wave
    const int n     = n0 + r15;
    const float bn  = bias ? bias[n] : 0.0f;

#pragma unroll
    for (int r = 0; r < 8; ++r) {
        float v = acc[r] + bn;
        if (do_relu) v = fmaxf(v, 0.0f);
        const size_t idx = (size_t)(mbase + r) * GCN_EMB + n;
        if (resid) v += resid[idx];
        out[idx] = v;
    }
}

// ---------------------------------------------------------------------------
// Edge scatter: agg[dst[e]] += h[src[e]]   (128 channels per edge)
// One wave per edge; each lane moves float4 (coalesced 512B gather per edge),
// then 4 global_atomic_add_f32 into the L2-resident agg buffer.
// ---------------------------------------------------------------------------
__global__ __launch_bounds__(256) void gcn_scatter_edges(
    const float* __restrict__ h,
    const int* __restrict__ src,
    const int* __restrict__ dst,
    float* __restrict__ agg,
    int n_edges)
{
    const long long gid  = (long long)blockIdx.x * blockDim.x + threadIdx.x;
    const int       e    = (int)(gid >> 5);
    const int       lane = (int)(gid & 31);
    if (e >= n_edges) return;

    const int s = src[e];
    const int d = dst[e];

    const float4 v = *(const float4*)(h + (size_t)s * GCN_EMB + lane * 4);
    float* o = agg + (size_t)d * GCN_EMB + lane * 4;
    atomicAdd(o + 0, v.x);
    atomicAdd(o + 1, v.y);
    atomicAdd(o + 2, v.z);
    atomicAdd(o + 3, v.w);
}

// ---------------------------------------------------------------------------
// Readout: out[node2graph[n]] += h[n]. One wave per node, float4 per lane.
// ---------------------------------------------------------------------------
__global__ __launch_bounds__(256) void gcn_readout_pool(
    const float* __restrict__ h,
    const int* __restrict__ node2graph,
    float* __restrict__ out,
    int n_nodes)
{
    const long long gid  = (long long)blockIdx.x * blockDim.x + threadIdx.x;
    const int       node = (int)(gid >> 5);
    const int       lane = (int)(gid & 31);
    if (node >= n_nodes) return;

    const int g = node2graph[node];
    const float4 v = *(const float4*)(h + (size_t)node * GCN_EMB + lane * 4);
    float* o = out + (size_t)g * GCN_EMB + lane * 4;
    atomicAdd(o + 0, v.x);
    atomicAdd(o + 1, v.y);
    atomicAdd(o + 2, v.z);
    atomicAdd(o + 3, v.w);
}

extern "C" void kernel_launch(void* const* d_in, const int* in_sizes, int n_in,
                              void* d_out, int out_size, void* d_ws, size_t ws_size,
                              hipStream_t stream) {
    (void)in_sizes; (void)n_in; (void)out_size; (void)ws_size;

    const float* x          = (const float*)d_in[0];
    const int*   src        = (const int*)  d_in[1];
    const int*   dst        = (const int*)  d_in[2];
    const int*   node2graph = (const int*)  d_in[3];
    const float* W_emb      = (const float*)d_in[4];
    const float* W0         = (const float*)d_in[5];
    const float* b0         = (const float*)d_in[6];
    const float* W1         = (const float*)d_in[7];
    const float* b1         = (const float*)d_in[8];
    const float* W2         = (const float*)d_in[9];
    const float* b2         = (const float*)d_in[10];
    float*       out        = (float*)d_out;

    const size_t hsz = (size_t)GCN_N_NODES * GCN_EMB;   // elements per node buffer
    float* hA  = (float*)d_ws;                          // ping
    float* hB  = hA + hsz;                              // pong
    float* agg = hB + hsz;                              // scatter accumulator

    const int gemm_blocks    = GCN_N_NODES / 16;                          // 3125 exact
    const int scatter_blocks = (GCN_N_EDGES * 32 + 255) / 256;            // wave per edge
    const int readout_blocks = (GCN_N_NODES * 32 + 255) / 256;            // wave per node
    const size_t hbytes  = hsz * sizeof(float);
    const size_t obytes  = (size_t)GCN_N_GRAPHS * GCN_EMB * sizeof(float);

    // h = x @ W_emb (no bias, no relu)
    gcn_gemm_n128_wmma<<<gemm_blocks, 256, 0, stream>>>(
        x, GCN_INP, W_emb, nullptr, nullptr, hA, 0);

    // Layer 0: agg = scatter(hA); hB = relu(agg@W0 + b0)
    hipMemsetAsync(agg, 0, hbytes, stream);
    gcn_scatter_edges<<<scatter_blocks, 256, 0, stream>>>(hA, src, dst, agg, GCN_N_EDGES);
    gcn_gemm_n128_wmma<<<gemm_blocks, 256, 0, stream>>>(
        agg, GCN_EMB, W0, b0, nullptr, hB, 1);

    // Layer 1: agg = scatter(hB); hA = relu(agg@W1 + b1) + hB
    hipMemsetAsync(agg, 0, hbytes, stream);
    gcn_scatter_edges<<<scatter_blocks, 256, 0, stream>>>(hB, src, dst, agg, GCN_N_EDGES);
    gcn_gemm_n128_wmma<<<gemm_blocks, 256, 0, stream>>>(
        agg, GCN_EMB, W1, b1, hB, hA, 1);

    // Layer 2: agg = scatter(hA); hB = relu(agg@W2 + b2) + hA
    hipMemsetAsync(agg, 0, hbytes, stream);
    gcn_scatter_edges<<<scatter_blocks, 256, 0, stream>>>(hA, src, dst, agg, GCN_N_EDGES);
    gcn_gemm_n128_wmma<<<gemm_blocks, 256, 0, stream>>>(
        agg, GCN_EMB, W2, b2, hA, hB, 1);

    // Readout: out[g] += hB[n]
    hipMemsetAsync(out, 0, obytes, stream);
    gcn_readout_pool<<<readout_blocks, 256, 0, stream>>>(hB, node2graph, out, GCN_N_NODES);
}